// KPlexPool_69569880260838
// MI455X (gfx1250) — compile-verified
//
#include <hip/hip_runtime.h>
#include <math.h>

#define Nn    100000
#define Ee    1600000
#define FEATc 128
#define Hc    128
#define C1c   50000
#define C2c   25000
#define Bc    64
#define NCLSc 10
#define NH    (Nn * Hc)   // 12.8M floats

typedef __attribute__((ext_vector_type(2))) float v2f;
typedef __attribute__((ext_vector_type(8))) float v8f;

static inline int cdiv(long long a, long long b) { return (int)((a + b - 1) / b); }

// ---------------- atomics helpers ----------------
__device__ __forceinline__ void atomAddF(float* p, float v) {
  __hip_atomic_fetch_add(p, v, __ATOMIC_RELAXED, __HIP_MEMORY_SCOPE_AGENT);
}
// valid only for non-negative floats (post-ReLU data): IEEE order == uint order
__device__ __forceinline__ void atomMaxNonNegF(float* p, float v) {
  __hip_atomic_fetch_max((unsigned int*)p, __float_as_uint(v),
                         __ATOMIC_RELAXED, __HIP_MEMORY_SCOPE_AGENT);
}
__device__ __forceinline__ void atomMaxI(int* p, int v) {
  __hip_atomic_fetch_max(p, v, __ATOMIC_RELAXED, __HIP_MEMORY_SCOPE_AGENT);
}

// ---------------- zero ----------------
__global__ void zero_f32(float* __restrict__ p, long long n) {
  long long i = (long long)blockIdx.x * blockDim.x + threadIdx.x;
  if (i < n) p[i] = 0.0f;
}

// ---------------- WMMA fp32 GEMM: C[M x 128] = A[M x K] @ B[K x 128] ----------------
// grid.x = ceil(M/16); block = 256 (8 waves); wave w -> column tile w.
// shared mem = 16*K*4 bytes.
__global__ void gemm128_wmma(const float* __restrict__ Amat,
                             const float* __restrict__ Bmat,
                             float* __restrict__ Cmat,
                             int M, int K) {
  extern __shared__ float lds[];                 // 16 x K tile of A
  const int m0  = blockIdx.x * 16;
  const int tid = threadIdx.x;

  for (int idx = tid; idx < 16 * K; idx += 256) {
    int r = idx / K, k = idx - r * K;
    int row = m0 + r;
    lds[idx] = (row < M) ? Amat[(size_t)row * K + k] : 0.0f;
  }
  __syncthreads();

  const int lane = tid & 31;
  const int wave = tid >> 5;          // 0..7
  const int half = lane >> 4;         // 0: lanes 0-15, 1: lanes 16-31
  const int l15  = lane & 15;
  const int col  = wave * 16 + l15;   // output column (< 128)

  v8f acc = {};
  for (int k0 = 0; k0 < K; k0 += 4) {
    const int ka = k0 + half * 2;     // ISA 16x4 layout: lane half offset = 2 in K
    v2f a, b;
    a.x = lds[l15 * K + ka];
    a.y = lds[l15 * K + ka + 1];
    b.x = Bmat[(size_t)ka * 128 + col];
    b.y = Bmat[(size_t)(ka + 1) * 128 + col];
    acc = __builtin_amdgcn_wmma_f32_16x16x4_f32(false, a, false, b,
                                                (short)0, acc, false, false);
  }

#pragma unroll
  for (int r = 0; r < 8; ++r) {       // C/D layout: VGPR r = rows r (lo half) / r+8 (hi half)
    int row = m0 + r + half * 8;
    if (row < M) Cmat[(size_t)row * 128 + col] = acc[r];
  }
}

// ---------------- graph kernels ----------------
__global__ void deg_accum(const int* __restrict__ dst, const float* __restrict__ w,
                          float* __restrict__ deg, int E) {
  int e = blockIdx.x * blockDim.x + threadIdx.x;
  if (e < E) atomAddF(&deg[dst[e]], w[e]);
}

__global__ void deg_to_dinv(float* __restrict__ deg, int n) {
  int i = blockIdx.x * blockDim.x + threadIdx.x;
  if (i < n) deg[i] = rsqrtf(deg[i] + 1.0f);
}

// one wave per edge; lane handles 4 consecutive features (128 feats / 32 lanes)
__global__ void gcn_agg_edges(const float* __restrict__ h,
                              const int* __restrict__ src,
                              const int* __restrict__ dst,
                              const float* __restrict__ w,
                              const float* __restrict__ dinv,
                              float* __restrict__ agg, int E) {
  int gwave = (blockIdx.x * blockDim.x + threadIdx.x) >> 5;
  int lane  = threadIdx.x & 31;
  if (gwave >= E) return;
  int s = src[gwave], d = dst[gwave];
  float nw = dinv[s] * w[gwave] * dinv[d];
  const float4 hv = ((const float4*)(h + (size_t)s * 128))[lane];
  float* ap = agg + (size_t)d * 128 + lane * 4;
  atomAddF(ap + 0, hv.x * nw);
  atomAddF(ap + 1, hv.y * nw);
  atomAddF(ap + 2, hv.z * nw);
  atomAddF(ap + 3, hv.w * nw);
}

// out = relu(agg + h*dinv^2 + bias)
__global__ void gcn_epilogue(const float* __restrict__ agg, const float* __restrict__ h,
                             const float* __restrict__ dinv, const float* __restrict__ bias,
                             float* __restrict__ out, int n) {
  long long i = (long long)blockIdx.x * blockDim.x + threadIdx.x;
  if (i >= (long long)n * Hc) return;
  int node = (int)(i >> 7), f = (int)(i & 127);
  float di = dinv[node];
  float v = agg[i] + h[i] * di * di + bias[f];
  out[i] = v > 0.0f ? v : 0.0f;
}

// accumulate segment sum + max directly into hcat[B, 768] at given column offsets
__global__ void pool_into_hcat(const float* __restrict__ x, const int* __restrict__ seg,
                               float* __restrict__ hcat, int n, int colSum, int colMax) {
  long long i = (long long)blockIdx.x * blockDim.x + threadIdx.x;
  if (i >= (long long)n * Hc) return;
  int node = (int)(i >> 7), f = (int)(i & 127);
  int b = seg[node];
  float v = x[i];
  atomAddF(&hcat[(size_t)b * 768 + colSum + f], v);
  atomMaxNonNegF(&hcat[(size_t)b * 768 + colMax + f], v);   // x is post-ReLU (>=0)
}

// cover_pool: xp[c, 0:128] = segment_sum, xp[c, 128:256] = segment_max
__global__ void pool_cluster(const float* __restrict__ x, const int* __restrict__ cluster,
                             float* __restrict__ xp, int n) {
  long long i = (long long)blockIdx.x * blockDim.x + threadIdx.x;
  if (i >= (long long)n * Hc) return;
  int node = (int)(i >> 7), f = (int)(i & 127);
  int c = cluster[node];
  float v = x[i];
  atomAddF(&xp[(size_t)c * 256 + f], v);
  atomMaxNonNegF(&xp[(size_t)c * 256 + 128 + f], v);        // x is post-ReLU (>=0)
}

__global__ void relabel_from(const int* __restrict__ srcO, const int* __restrict__ dstO,
                             const int* __restrict__ cl, int* __restrict__ srcR,
                             int* __restrict__ dstR, int E) {
  int e = blockIdx.x * blockDim.x + threadIdx.x;
  if (e < E) { srcR[e] = cl[srcO[e]]; dstR[e] = cl[dstO[e]]; }
}

__global__ void relabel_inplace(int* __restrict__ srcR, int* __restrict__ dstR,
                                const int* __restrict__ cl, int E) {
  int e = blockIdx.x * blockDim.x + threadIdx.x;
  if (e < E) { srcR[e] = cl[srcR[e]]; dstR[e] = cl[dstR[e]]; }
}

__global__ void seg_max_int(const int* __restrict__ vals, const int* __restrict__ seg,
                            int* __restrict__ out, int n) {
  int i = blockIdx.x * blockDim.x + threadIdx.x;
  if (i < n) atomMaxI(&out[seg[i]], vals[i]);   // vals >= 0, out zero-initialized
}

__global__ void count_seg(const int* __restrict__ seg, float* __restrict__ cnt, int n) {
  int i = blockIdx.x * blockDim.x + threadIdx.x;
  if (i < n) atomAddF(&cnt[seg[i]], 1.0f);
}

__global__ void mean_div(float* __restrict__ hcat, const float* __restrict__ cnt1,
                         const float* __restrict__ cnt2) {
  int i = blockIdx.x * blockDim.x + threadIdx.x;   // B*128
  if (i >= Bc * Hc) return;
  int b = i >> 7, f = i & 127;
  hcat[(size_t)b * 768 + 256 + f] /= cnt1[b];
  hcat[(size_t)b * 768 + 512 + f] /= cnt2[b];
}

__global__ void bias_relu(const float* __restrict__ in, const float* __restrict__ bias,
                          float* __restrict__ out, int nelem) {
  int i = blockIdx.x * blockDim.x + threadIdx.x;
  if (i >= nelem) return;
  float v = in[i] + bias[i & 127];
  out[i] = v > 0.0f ? v : 0.0f;
}

// logits = hmid @ lw2 + lb2 ; softmax ; one thread per graph
__global__ void head_kernel(const float* __restrict__ hmid, const float* __restrict__ lw2,
                            const float* __restrict__ lb2, float* __restrict__ out) {
  int b = blockIdx.x * blockDim.x + threadIdx.x;
  if (b >= Bc) return;
  float lg[NCLSc], mx = -1e30f;
  for (int c = 0; c < NCLSc; ++c) {
    float s = lb2[c];
    for (int k = 0; k < Hc; ++k) s += hmid[b * Hc + k] * lw2[k * NCLSc + c];
    lg[c] = s; if (s > mx) mx = s;
  }
  float den = 0.0f;
  for (int c = 0; c < NCLSc; ++c) { lg[c] = expf(lg[c] - mx); den += lg[c]; }
  for (int c = 0; c < NCLSc; ++c) out[b * NCLSc + c] = lg[c] / den;
}

// ---------------- orchestration ----------------
extern "C" void kernel_launch(void* const* d_in, const int* in_sizes, int n_in,
                              void* d_out, int out_size, void* d_ws, size_t ws_size,
                              hipStream_t stream) {
  (void)in_sizes; (void)n_in; (void)out_size; (void)ws_size;
  const float* x        = (const float*)d_in[0];
  const int*   srcO     = (const int*)d_in[1];          // edge_index row 0
  const int*   dstO     = srcO + Ee;                    // edge_index row 1
  const float* w        = (const float*)d_in[2];
  const int*   batch    = (const int*)d_in[3];
  const int*   cluster1 = (const int*)d_in[4];
  const int*   cluster2 = (const int*)d_in[5];
  const float* W_in = (const float*)d_in[6];
  const float* b_in = (const float*)d_in[7];
  const float* W1   = (const float*)d_in[8];
  const float* b1   = (const float*)d_in[9];
  const float* W2   = (const float*)d_in[10];
  const float* b2   = (const float*)d_in[11];
  const float* lw1  = (const float*)d_in[12];
  const float* lb1  = (const float*)d_in[13];
  const float* lw2  = (const float*)d_in[14];
  const float* lb2  = (const float*)d_in[15];
  float* out = (float*)d_out;

  // workspace layout
  float* ws      = (float*)d_ws;
  float* bufH    = ws;                    // GEMM output h          (N*H)
  float* bufX    = bufH   + NH;           // x1 / x2 / x3           (N*H)
  float* bufXP   = bufX   + NH;           // pooled xp              (C1*2H max = N*H)
  float* bufAgg  = bufXP  + NH;           // scatter-add target     (N*H)
  float* degb    = bufAgg + NH;           // degree / dinv          (N)
  int*   srcR    = (int*)(degb + Nn);     // relabeled src          (E)
  int*   dstR    = srcR + Ee;             // relabeled dst          (E)
  int*   batch1  = dstR + Ee;             // (C1)
  int*   batch2  = batch1 + C1c;          // (C2)
  float* cnt1    = (float*)(batch2 + C2c);
  float* cnt2    = cnt1 + Bc;
  float* hcat    = cnt2 + Bc;             // [B, 768]
  float* gmid    = hcat + (size_t)Bc * 6 * Hc; // raw head GEMM out [B,128]
  float* hmid    = gmid + Bc * Hc;

  const int T = 256;
  #define Z(p, n) zero_f32<<<cdiv((long long)(n), T), T, 0, stream>>>((float*)(p), (long long)(n))

  // ================= Stage 0: GCN on N nodes, K=128 =================
  gemm128_wmma<<<cdiv(Nn, 16), T, 16 * FEATc * sizeof(float), stream>>>(x, W_in, bufH, Nn, FEATc);
  Z(degb, Nn); Z(bufAgg, (long long)Nn * Hc);
  deg_accum<<<cdiv(Ee, T), T, 0, stream>>>(dstO, w, degb, Ee);
  deg_to_dinv<<<cdiv(Nn, T), T, 0, stream>>>(degb, Nn);
  gcn_agg_edges<<<cdiv((long long)Ee * 32, T), T, 0, stream>>>(bufH, srcO, dstO, w, degb, bufAgg, Ee);
  gcn_epilogue<<<cdiv((long long)Nn * Hc, T), T, 0, stream>>>(bufAgg, bufH, degb, b_in, bufX, Nn);

  Z(hcat, (long long)Bc * 6 * Hc);
  pool_into_hcat<<<cdiv((long long)Nn * Hc, T), T, 0, stream>>>(bufX, batch, hcat, Nn, 0, 128);

  // ================= Pool stage 1 (N -> C1) =================
  Z(bufXP, (long long)C1c * 2 * Hc);
  pool_cluster<<<cdiv((long long)Nn * Hc, T), T, 0, stream>>>(bufX, cluster1, bufXP, Nn);
  relabel_from<<<cdiv(Ee, T), T, 0, stream>>>(srcO, dstO, cluster1, srcR, dstR, Ee);
  Z(batch1, C1c);
  seg_max_int<<<cdiv(Nn, T), T, 0, stream>>>(batch, cluster1, batch1, Nn);

  // GCN on C1 clusters, K=256
  gemm128_wmma<<<cdiv(C1c, 16), T, 16 * 256 * sizeof(float), stream>>>(bufXP, W1, bufH, C1c, 256);
  Z(degb, C1c); Z(bufAgg, (long long)C1c * Hc);
  deg_accum<<<cdiv(Ee, T), T, 0, stream>>>(dstR, w, degb, Ee);
  deg_to_dinv<<<cdiv(C1c, T), T, 0, stream>>>(degb, C1c);
  gcn_agg_edges<<<cdiv((long long)Ee * 32, T), T, 0, stream>>>(bufH, srcR, dstR, w, degb, bufAgg, Ee);
  gcn_epilogue<<<cdiv((long long)C1c * Hc, T), T, 0, stream>>>(bufAgg, bufH, degb, b1, bufX, C1c); // x2

  Z(cnt1, Bc);
  count_seg<<<cdiv(C1c, T), T, 0, stream>>>(batch1, cnt1, C1c);
  pool_into_hcat<<<cdiv((long long)C1c * Hc, T), T, 0, stream>>>(bufX, batch1, hcat, C1c, 256, 384);

  // ================= Pool stage 2 (C1 -> C2) =================
  Z(bufXP, (long long)C2c * 2 * Hc);
  pool_cluster<<<cdiv((long long)C1c * Hc, T), T, 0, stream>>>(bufX, cluster2, bufXP, C1c);
  relabel_inplace<<<cdiv(Ee, T), T, 0, stream>>>(srcR, dstR, cluster2, Ee);
  Z(batch2, C2c);
  seg_max_int<<<cdiv(C1c, T), T, 0, stream>>>(batch1, cluster2, batch2, C1c);

  // GCN on C2 clusters, K=256
  gemm128_wmma<<<cdiv(C2c, 16), T, 16 * 256 * sizeof(float), stream>>>(bufXP, W2, bufH, C2c, 256);
  Z(degb, C2c); Z(bufAgg, (long long)C2c * Hc);
  deg_accum<<<cdiv(Ee, T), T, 0, stream>>>(dstR, w, degb, Ee);
  deg_to_dinv<<<cdiv(C2c, T), T, 0, stream>>>(degb, C2c);
  gcn_agg_edges<<<cdiv((long long)Ee * 32, T), T, 0, stream>>>(bufH, srcR, dstR, w, degb, bufAgg, Ee);
  gcn_epilogue<<<cdiv((long long)C2c * Hc, T), T, 0, stream>>>(bufAgg, bufH, degb, b2, bufX, C2c); // x3

  Z(cnt2, Bc);
  count_seg<<<cdiv(C2c, T), T, 0, stream>>>(batch2, cnt2, C2c);
  pool_into_hcat<<<cdiv((long long)C2c * Hc, T), T, 0, stream>>>(bufX, batch2, hcat, C2c, 512, 640);

  // ================= Head =================
  mean_div<<<cdiv(Bc * Hc, T), T, 0, stream>>>(hcat, cnt1, cnt2);
  gemm128_wmma<<<cdiv(Bc, 16), T, 16 * 768 * sizeof(float), stream>>>(hcat, lw1, gmid, Bc, 768);
  bias_relu<<<cdiv(Bc * Hc, T), T, 0, stream>>>(gmid, lb1, hmid, Bc * Hc);
  head_kernel<<<2, 32, 0, stream>>>(hmid, lw2, lb2, out);

  #undef Z
}